// RelConvLayer_56487409877774
// MI455X (gfx1250) — compile-verified
//
#include <hip/hip_runtime.h>

typedef __attribute__((ext_vector_type(2))) float v2f;
typedef __attribute__((ext_vector_type(4))) float v4f;
typedef __attribute__((ext_vector_type(8))) float v8f;

#define DD 128  // embedding dim, fixed by the reference

// ---------------------------------------------------------------------------
// 1. zero fill
// ---------------------------------------------------------------------------
__global__ void zero_f32_kernel(float* p, long n) {
  long i = (long)blockIdx.x * blockDim.x + threadIdx.x;
  long stride = (long)gridDim.x * blockDim.x;
  for (; i < n; i += stride) p[i] = 0.0f;
}

// ---------------------------------------------------------------------------
// 2. degree counting: deg = segment_sum(ones, edge_index[0]) per half
// ---------------------------------------------------------------------------
__global__ void degree_kernel(const int* __restrict__ ei, int E, int E2,
                              float* __restrict__ deg_in,
                              float* __restrict__ deg_out) {
  int e = blockIdx.x * blockDim.x + threadIdx.x;
  if (e >= E2) return;
  int row = ei[e];  // edge_index[0][e]
  float* deg = (e < E) ? deg_in : deg_out;
#if defined(__HIP_DEVICE_COMPILE__)
  unsafeAtomicAdd(&deg[row], 1.0f);
#else
  atomicAdd(&deg[row], 1.0f);
#endif
}

// ---------------------------------------------------------------------------
// 3. T = A(M x 128) @ B(128 x 128) via V_WMMA_F32_16X16X4_F32 (exact f32)
//    one wave (32 threads) per 16x16 output tile; K-loop fully unrolled.
//    A 16x4 layout: lane l -> M = l%16, K = (l>>4)*2 + {0,1} in vgpr {x,y}
//    B 4x16 / C,D 16x16: row striped across lanes within a VGPR.
//    Rows >= M are clamped to row M-1 (computed but never stored) so the
//    inner loop has no divergent guards -> pure loads + WMMA.
// ---------------------------------------------------------------------------
__global__ void rel_gemm_wmma_kernel(const float* __restrict__ A_in,
                                     const float* __restrict__ B_in,
                                     const float* __restrict__ A_out,
                                     const float* __restrict__ B_out,
                                     float* __restrict__ T_in,
                                     float* __restrict__ T_out, int M) {
  int MT = (M + 15) >> 4;  // tiles in M
  int tiles_per_mat = MT * (DD / 16);
  int tile = blockIdx.x;
  const float *A, *B;
  float* T;
  if (tile < tiles_per_mat) {
    A = A_in; B = B_in; T = T_in;
  } else {
    tile -= tiles_per_mat;
    A = A_out; B = B_out; T = T_out;
  }
  int mt = tile / (DD / 16);
  int nt = tile % (DD / 16);
  int m0 = mt << 4;
  int n0 = nt << 4;

  int l    = threadIdx.x & 31;
  int half = l >> 4;  // 0: lanes 0-15, 1: lanes 16-31
  int lm   = l & 15;

  int mrow = min(m0 + lm, M - 1);  // clamp: no divergent guard on loads
  const float* arow = A + (long)mrow * DD;
  const float* bcol = B + n0 + lm;

  v8f c = {};
#pragma unroll
  for (int k0 = 0; k0 < DD; k0 += 4) {
    int ka = k0 + half * 2;
    v2f a = *(const v2f*)(arow + ka);  // one b64 load (K=ka, ka+1)
    v2f b;
    b.x = bcol[(long)ka * DD];
    b.y = bcol[(long)(ka + 1) * DD];
    c = __builtin_amdgcn_wmma_f32_16x16x4_f32(
        /*neg_a=*/false, a, /*neg_b=*/false, b,
        /*c_mod=*/(short)0, c, /*reuse_a=*/false, /*reuse_b=*/false);
  }

#pragma unroll
  for (int r = 0; r < 8; ++r) {
    int mr = m0 + r + half * 8;  // C/D: vgpr r holds M=r (lanes<16) / M=r+8
    if (mr < M) T[(long)mr * DD + n0 + lm] = c[r];
  }
}

// ---------------------------------------------------------------------------
// 4. norm-weighted scatter: one wave32 per edge; each lane owns 4 contiguous
//    dims -> one global_load_b128 + 4 global_atomic_add_f32.
// ---------------------------------------------------------------------------
__global__ void scatter_kernel(const int* __restrict__ ei,
                               const int* __restrict__ et, int E, int E2,
                               const float* __restrict__ deg_in,
                               const float* __restrict__ deg_out,
                               const float* __restrict__ T_in,
                               const float* __restrict__ T_out,
                               float* __restrict__ h) {
  long gtid = (long)blockIdx.x * blockDim.x + threadIdx.x;
  int wave = (int)(gtid >> 5);
  int lane = threadIdx.x & 31;
  if (wave >= E2) return;

  bool first_half = (wave < E);
  const float* deg = first_half ? deg_in : deg_out;
  const float* T   = first_half ? T_in : T_out;

  int row = ei[wave];       // edge_index[0][wave]
  int col = ei[E2 + wave];  // edge_index[1][wave]
  int t   = et[wave];

  float dr = deg[row];
  float dc = deg[col];
  float ninv = ((dr > 0.0f) ? __frsqrt_rn(dr) : 0.0f) *
               ((dc > 0.0f) ? __frsqrt_rn(dc) : 0.0f);
  float w = 0.5f * ninv;

  int d0 = lane * 4;  // 4 contiguous dims per lane
  const float* src = T + (long)t * DD + d0;
  float* dst = h + (long)row * DD + d0;

  v4f v = *(const v4f*)src;  // one b128 load
#pragma unroll
  for (int j = 0; j < 4; ++j) {
#if defined(__HIP_DEVICE_COMPILE__)
    unsafeAtomicAdd(&dst[j], w * v[j]);
#else
    atomicAdd(&dst[j], w * v[j]);
#endif
  }
}

// ---------------------------------------------------------------------------
// 5. per-column sum / sum-of-squares (thread = column)
// ---------------------------------------------------------------------------
__global__ void bn_reduce_kernel(const float* __restrict__ h, int num_ent,
                                 float* __restrict__ sums,
                                 float* __restrict__ sumsq) {
  int colid = threadIdx.x;  // 128 threads
  int r0 = blockIdx.x * 256;
  int rend = min(r0 + 256, num_ent);
  float s = 0.0f, s2 = 0.0f;
  for (int r = r0; r < rend; ++r) {
    float v = h[(long)r * DD + colid];
    s += v;
    s2 += v * v;
  }
#if defined(__HIP_DEVICE_COMPILE__)
  unsafeAtomicAdd(&sums[colid], s);
  unsafeAtomicAdd(&sumsq[colid], s2);
#else
  atomicAdd(&sums[colid], s);
  atomicAdd(&sumsq[colid], s2);
#endif
}

// ---------------------------------------------------------------------------
// 6. batchnorm (biased var) + tanh, in place in d_out
// ---------------------------------------------------------------------------
__global__ void bn_tanh_kernel(float* __restrict__ h, int num_ent,
                               const float* __restrict__ sums,
                               const float* __restrict__ sumsq,
                               const float* __restrict__ gamma,
                               const float* __restrict__ beta) {
  long n = (long)num_ent * DD;
  long i = (long)blockIdx.x * blockDim.x + threadIdx.x;
  long stride = (long)gridDim.x * blockDim.x;
  float invN = 1.0f / (float)num_ent;
  for (; i < n; i += stride) {
    int c = (int)(i & (DD - 1));
    float mean = sums[c] * invN;
    float var = sumsq[c] * invN - mean * mean;
    float v = (h[i] - mean) * __frsqrt_rn(var + 1e-5f) * gamma[c] + beta[c];
    h[i] = tanhf(v);
  }
}

// ---------------------------------------------------------------------------
extern "C" void kernel_launch(void* const* d_in, const int* in_sizes, int n_in,
                              void* d_out, int out_size, void* d_ws,
                              size_t ws_size, hipStream_t stream) {
  const float* rel_embed     = (const float*)d_in[0];
  const float* rel_embed_in  = (const float*)d_in[1];
  const float* rel_embed_out = (const float*)d_in[2];
  const float* w_in          = (const float*)d_in[3];
  const float* w_out         = (const float*)d_in[4];
  const float* bn_gamma      = (const float*)d_in[5];
  const float* bn_beta       = (const float*)d_in[6];
  const int*   edge_index    = (const int*)d_in[7];
  const int*   edge_type     = (const int*)d_in[8];
  // d_in[9] = num_ent on device; derive on host instead (no sync reads):
  const int NUM_REL = in_sizes[0] / DD;                // 500
  const int E2      = in_sizes[8];                     // 3,000,000
  const int E       = E2 / 2;                          // 1,500,000
  const int num_ent = (out_size - NUM_REL * DD) / DD;  // 200,000

  float* h = (float*)d_out;  // accumulate h directly in the output buffer

  // workspace layout: [deg_in | deg_out | sums | sumsq | T_in | T_out]
  float* deg_in  = (float*)d_ws;
  float* deg_out = deg_in + num_ent;
  float* sums    = deg_out + num_ent;
  float* sumsq   = sums + DD;
  float* T_in    = sumsq + DD;
  float* T_out   = T_in + (long)NUM_REL * DD;

  // 1. zero h and the zero-initialized workspace region (deg + sums)
  zero_f32_kernel<<<2048, 256, 0, stream>>>(h, (long)num_ent * DD);
  zero_f32_kernel<<<512, 256, 0, stream>>>(deg_in, 2L * num_ent + 2 * DD);

  // 2. degrees
  degree_kernel<<<(E2 + 255) / 256, 256, 0, stream>>>(edge_index, E, E2,
                                                      deg_in, deg_out);

  // 3. relation-table GEMMs via WMMA (one wave per 16x16 tile)
  {
    int MT = (NUM_REL + 15) / 16;
    int blocks = 2 * MT * (DD / 16);
    rel_gemm_wmma_kernel<<<blocks, 32, 0, stream>>>(
        rel_embed_in, w_in, rel_embed_out, w_out, T_in, T_out, NUM_REL);
  }

  // 4. scatter-add (one wave per edge)
  {
    long total_threads = (long)E2 * 32;
    int blocks = (int)((total_threads + 255) / 256);
    scatter_kernel<<<blocks, 256, 0, stream>>>(edge_index, edge_type, E, E2,
                                               deg_in, deg_out, T_in, T_out, h);
  }

  // 5. BN statistics
  bn_reduce_kernel<<<(num_ent + 255) / 256, 128, 0, stream>>>(h, num_ent, sums,
                                                              sumsq);

  // 6. BN + tanh in place
  bn_tanh_kernel<<<4096, 256, 0, stream>>>(h, num_ent, sums, sumsq, bn_gamma,
                                           bn_beta);

  // 7. second output: rel_embed passthrough
  hipMemcpyAsync((float*)d_out + (long)num_ent * DD, rel_embed,
                 (size_t)in_sizes[0] * sizeof(float), hipMemcpyDeviceToDevice,
                 stream);
}